// BitfieldLinear_53876069761217
// MI455X (gfx1250) — compile-verified
//
#include <hip/hip_runtime.h>
#include <math.h>

// ---------------------------------------------------------------------------
// BitfieldLinear on MI455X (gfx1250):
//   y = x @ W^T,  W[o,:] = sign(o)*tanh(r(o)) * basis[idx(o),:]
// Factorized:  T = x @ basis^T  ([128,256] f32 GEMM via V_WMMA_F32_16X16X4_F32)
//              y[b,o] = scale[o] * T[b, idx[o]]   (decode + gather + scale)
// ---------------------------------------------------------------------------

#define IN_FEATURES 4096
#define OUT_FEATURES 8192
#define BASIS_SIZE 256
#define BATCH 128

typedef __attribute__((ext_vector_type(2))) float v2f;
typedef __attribute__((ext_vector_type(8))) float v8f;

// ---------------------------------------------------------------------------
// Phase 1: T[128,256] = x[128,4096] @ basis[256,4096]^T
// One wave32 computes one 16x16 tile of T. 128 tiles total.
//
// f32 WMMA operand layout (CDNA5 ISA 7.12.2, 32-bit A 16x4):
//   A: lane L<16  -> VGPR0 = A[M=L][k0+0], VGPR1 = A[M=L][k0+1]
//      lane 16+L -> VGPR0 = A[M=L][k0+2], VGPR1 = A[M=L][k0+3]
//   => each lane loads one contiguous float2 from x per WMMA step.
//   B (4x16, K x N): B[kk][n] = basis[nbase+n][k0+kk]; same half-wave K split
//   => each lane loads one contiguous float2 from basis per WMMA step.
//   C/D 16x16 f32: VGPR v -> M=v (lanes 0-15), M=8+v (lanes 16-31), N=lane&15.
// ---------------------------------------------------------------------------
__global__ __launch_bounds__(256) void bitfield_gemm_T(
    const float* __restrict__ x,      // [BATCH, IN_FEATURES]
    const float* __restrict__ basis,  // [BASIS_SIZE, IN_FEATURES]
    float* __restrict__ T)            // [BATCH, BASIS_SIZE]
{
    const int wave = threadIdx.x >> 5;                 // 8 waves / block
    const int lane = threadIdx.x & 31;
    const int tile = blockIdx.x * 8 + wave;            // 0..127
    const int tiles_k = BASIS_SIZE / 16;               // 16
    const int tb = tile / tiles_k;                     // batch tile 0..7
    const int tk = tile % tiles_k;                     // basis tile 0..15

    const int mrow  = lane & 15;                       // M (for A) / N (for B)
    const int khalf = (lane >> 4) << 1;                // 0 or 2 (K sub-offset)

    const float* xrow = x     + (size_t)(tb * 16 + mrow) * IN_FEATURES + khalf;
    const float* brow = basis + (size_t)(tk * 16 + mrow) * IN_FEATURES + khalf;

    // 4 independent accumulators to break the WMMA C-chain RAW latency.
    v8f acc0 = {}; v8f acc1 = {}; v8f acc2 = {}; v8f acc3 = {};

    for (int k = 0; k < IN_FEATURES; k += 16) {
        v2f a0 = *(const v2f*)(xrow + k +  0);
        v2f b0 = *(const v2f*)(brow + k +  0);
        v2f a1 = *(const v2f*)(xrow + k +  4);
        v2f b1 = *(const v2f*)(brow + k +  4);
        v2f a2 = *(const v2f*)(xrow + k +  8);
        v2f b2 = *(const v2f*)(brow + k +  8);
        v2f a3 = *(const v2f*)(xrow + k + 12);
        v2f b3 = *(const v2f*)(brow + k + 12);
        acc0 = __builtin_amdgcn_wmma_f32_16x16x4_f32(false, a0, false, b0,
                                                     (short)0, acc0, false, false);
        acc1 = __builtin_amdgcn_wmma_f32_16x16x4_f32(false, a1, false, b1,
                                                     (short)0, acc1, false, false);
        acc2 = __builtin_amdgcn_wmma_f32_16x16x4_f32(false, a2, false, b2,
                                                     (short)0, acc2, false, false);
        acc3 = __builtin_amdgcn_wmma_f32_16x16x4_f32(false, a3, false, b3,
                                                     (short)0, acc3, false, false);
    }
    v8f acc = (acc0 + acc1) + (acc2 + acc3);

    // D layout store
    const int n      = lane & 15;
    const int mbase  = (lane >> 4) * 8;
    float* out = T + (size_t)(tb * 16 + mbase) * BASIS_SIZE + tk * 16 + n;
#pragma unroll
    for (int v = 0; v < 8; ++v) {
        out[(size_t)v * BASIS_SIZE] = acc[v];
    }
}

// ---------------------------------------------------------------------------
// Phase 2: decode 21-bit codes, gather from T, scale, write y.
// One thread owns one output column o (decodes code + tanhf exactly once),
// loops over a 16-batch chunk. Stores are fully coalesced; the T[b,idx]
// gather hits a hot 1 KB row (T is 128 KB, L2-resident).
// ---------------------------------------------------------------------------
#define B_CHUNK 16

__global__ __launch_bounds__(256) void bitfield_decode_gather(
    const int* __restrict__ codes,    // [OUT_FEATURES]
    const float* __restrict__ T,      // [BATCH, BASIS_SIZE]
    float* __restrict__ y)            // [BATCH, OUT_FEATURES]
{
    const int o  = blockIdx.x * blockDim.x + threadIdx.x;  // 0..8191
    const int b0 = blockIdx.y * B_CHUNK;

    const int code = codes[o];
    const int idx  = code & 0xFF;                 // bits [0:8)
    const int rq   = (code >> 8) & 0xFFF;         // bits [8:20)
    float s = tanhf((float)rq * (1.0f / 4095.0f));
    if (code & (1 << 20)) s = -s;                 // bit 20 = sign

    const float* trow = T + (size_t)b0 * BASIS_SIZE + idx;
    float* yrow = y + (size_t)b0 * OUT_FEATURES + o;
#pragma unroll
    for (int bb = 0; bb < B_CHUNK; ++bb) {
        yrow[(size_t)bb * OUT_FEATURES] = s * trow[(size_t)bb * BASIS_SIZE];
    }
}

// ---------------------------------------------------------------------------
// Host launcher. setup_inputs order: x, codes, basis. Output f32 [128, 8192].
// d_ws: first 128 KB used for T.
// ---------------------------------------------------------------------------
extern "C" void kernel_launch(void* const* d_in, const int* in_sizes, int n_in,
                              void* d_out, int out_size, void* d_ws, size_t ws_size,
                              hipStream_t stream) {
    (void)in_sizes; (void)n_in; (void)out_size; (void)ws_size;
    const float* x     = (const float*)d_in[0];
    const int*   codes = (const int*)d_in[1];
    const float* basis = (const float*)d_in[2];
    float* y = (float*)d_out;
    float* T = (float*)d_ws;  // 128*256*4 = 128 KB scratch

    // Phase 1: 128 tiles, 8 waves (256 threads) per block -> 16 blocks.
    bitfield_gemm_T<<<dim3(16), dim3(256), 0, stream>>>(x, basis, T);

    // Phase 2: grid (8192/256 columns, 128/16 batch-chunks).
    bitfield_decode_gather<<<dim3(OUT_FEATURES / 256, BATCH / B_CHUNK),
                             dim3(256), 0, stream>>>(codes, T, y);
}